// WindowAttention_29807073034950
// MI455X (gfx1250) — compile-verified
//
#include <hip/hip_runtime.h>
#include <hip/hip_bf16.h>

typedef float v2f __attribute__((ext_vector_type(2)));
typedef float v8f __attribute__((ext_vector_type(8)));

#define NTOK   49
#define NP     64
#define DIM    256
#define HEADS  8
#define HD     32
#define WINE   (DIM * NTOK)        /* 12544 floats per window */
#define TOK_S  257
#define QKV_S  33
#define STG_S  65
#define AO_S   257
#define QSCALE 0.17677669529663687f   /* 32^-0.5 */

/* LDS layout (float offsets) */
#define TOK_OFF 0
#define TOK_SZ  (NP * TOK_S)                 /* 16448 */
#define Q_OFF   (TOK_OFF + TOK_SZ)           /* 16448 */
#define PART_SZ (HEADS * NP * QKV_S)         /* 16896 */
#define K_OFF   (Q_OFF + PART_SZ)            /* 33344 */
#define V_OFF   (K_OFF + PART_SZ)            /* 50240 */
#define BIAS_OFF (V_OFF + PART_SZ)           /* 67136 */
#define BIAS_SZ  (169 * HEADS)               /* 1352  */
#define LDS_FLOATS (BIAS_OFF + BIAS_SZ)      /* 68488 floats = 273952 B */
#define STG_SZ  (NP * STG_S)                 /* 4160; 8 waves * 4160 = 33280 <= K_OFF */
#define AO_OFF  K_OFF                        /* reuse dead k region for attn-out */

/* ---- fp32 WMMA fragment loaders (16x16x4) ----
 * A 16x4 (MxK): lanes 0-15 -> M=m0+lane, K=k0+{0,1}; lanes 16-31 -> K=k0+{2,3}
 * B  4x16 (KxN): lanes 0-15 -> N=n0+lane, K rows striped the same way
 * C 16x16: VGPR g -> M = g (+8 for upper half), N = lane&15
 */
__device__ __forceinline__ v2f fragA(const float* p, int stride, int m0, int k0,
                                     int l15, int khh) {
  const float* q = p + (m0 + l15) * stride + (k0 + khh);
  v2f r; r.x = q[0]; r.y = q[1]; return r;
}
__device__ __forceinline__ v2f fragB_KN(const float* p, int stride, int k0, int n0,
                                        int l15, int khh) {
  const float* q = p + (k0 + khh) * stride + (n0 + l15);
  v2f r; r.x = q[0]; r.y = q[stride]; return r;
}
__device__ __forceinline__ v2f fragB_NK(const float* p, int stride, int k0, int n0,
                                        int l15, int khh) {
  const float* q = p + (n0 + l15) * stride + (k0 + khh);
  v2f r; r.x = q[0]; r.y = q[1]; return r;
}

#define WMMA_F32(acc, a, b) \
  (acc) = __builtin_amdgcn_wmma_f32_16x16x4_f32(false, (a), false, (b), (short)0, (acc), false, false)

__global__ __launch_bounds__(256)
void win_attn_fused(const float* __restrict__ x,
                    const float* __restrict__ Wqkv,
                    const float* __restrict__ Wout,
                    const float* __restrict__ bt,
                    float* __restrict__ out) {
  extern __shared__ float smem[];
  const int tid  = threadIdx.x;
  const int wv   = tid >> 5;          /* wave id 0..7 == head id */
  const int lane = tid & 31;
  const int l15  = lane & 15;
  const int khh  = (lane >> 4) << 1;  /* K offset 0 or 2 for A/B frags */
  const int mh   = (lane >> 4) << 3;  /* M offset 0 or 8 for C tiles */
  const int win  = blockIdx.x;

  const v8f vzero = {0.f,0.f,0.f,0.f,0.f,0.f,0.f,0.f};

  /* ---- Stage 0: async-copy tokens straight into LDS with (d,n)->(n,d)
   *      transpose scatter (per-lane LDS dest addresses), zero-pad rows
   *      49..63, and stage the tiny bias table in LDS. ---- */
  {
    const float* xw = x + (size_t)win * WINE;
    for (int idx = tid; idx < WINE; idx += 256) {     /* 49 iters, all lanes live */
      int c = idx / NTOK;
      int i = idx - c * NTOK;
      unsigned ldsb = (unsigned)((TOK_OFF + i * TOK_S + c) * sizeof(float));
      const float* g = xw + idx;
      asm volatile("global_load_async_to_lds_b32 %0, %1, off"
                   :: "v"(ldsb), "v"(g) : "memory");
    }
    for (int idx = tid; idx < (NP - NTOK) * DIM; idx += 256) {
      int i = NTOK + idx / DIM;
      int c = idx & (DIM - 1);
      smem[TOK_OFF + i * TOK_S + c] = 0.0f;
    }
    for (int idx = tid; idx < BIAS_SZ; idx += 256)
      smem[BIAS_OFF + idx] = bt[idx];
    asm volatile("s_wait_asynccnt 0x0" ::: "memory");
  }
  __syncthreads();

  /* ---- Stage 1: QKV = tok(64x256) @ Wqkv(256x768); wave owns 96 cols ---- */
  {
    const int nb = wv * 96;
    v8f acc[4][6];
#pragma unroll
    for (int mt = 0; mt < 4; ++mt)
#pragma unroll
      for (int nt = 0; nt < 6; ++nt) acc[mt][nt] = vzero;

#pragma unroll 1
    for (int kk = 0; kk < DIM; kk += 4) {
      v2f a[4], b[6];
#pragma unroll
      for (int mt = 0; mt < 4; ++mt)
        a[mt] = fragA(smem + TOK_OFF, TOK_S, mt * 16, kk, l15, khh);
#pragma unroll
      for (int nt = 0; nt < 6; ++nt)
        b[nt] = fragB_KN(Wqkv, 3 * DIM, kk, nb + nt * 16, l15, khh);
#pragma unroll
      for (int mt = 0; mt < 4; ++mt)
#pragma unroll
        for (int nt = 0; nt < 6; ++nt)
          WMMA_F32(acc[mt][nt], a[mt], b[nt]);
    }
    /* scatter to LDS [part][head][row][ch]; pre-scale q */
#pragma unroll
    for (int mt = 0; mt < 4; ++mt)
#pragma unroll
      for (int nt = 0; nt < 6; ++nt)
#pragma unroll
        for (int g = 0; g < 8; ++g) {
          int m    = mt * 16 + g + mh;
          int n    = nb + nt * 16 + l15;
          int part = n >> 8;
          int c    = n & (DIM - 1);
          int hh   = c >> 5;
          int ch   = c & (HD - 1);
          float val = acc[mt][nt][g];
          if (part == 0) val *= QSCALE;
          smem[Q_OFF + part * PART_SZ + hh * (NP * QKV_S) + m * QKV_S + ch] = val;
        }
  }
  __syncthreads();

  /* ---- Stage 2: S = q @ k^T + bias, masked softmax (head = wv) ---- */
  v8f sacc[4][4];
  {
#pragma unroll
    for (int mt = 0; mt < 4; ++mt)
#pragma unroll
      for (int nt = 0; nt < 4; ++nt) sacc[mt][nt] = vzero;

    const float* qh = smem + Q_OFF + wv * (NP * QKV_S);
    const float* kh = smem + K_OFF + wv * (NP * QKV_S);
#pragma unroll 1
    for (int kk = 0; kk < HD; kk += 4) {
      v2f a[4], b[4];
#pragma unroll
      for (int mt = 0; mt < 4; ++mt) a[mt] = fragA(qh, QKV_S, mt * 16, kk, l15, khh);
#pragma unroll
      for (int nt = 0; nt < 4; ++nt) b[nt] = fragB_NK(kh, QKV_S, kk, nt * 16, l15, khh);
#pragma unroll
      for (int mt = 0; mt < 4; ++mt)
#pragma unroll
        for (int nt = 0; nt < 4; ++nt)
          WMMA_F32(sacc[mt][nt], a[mt], b[nt]);
    }

    /* softmax per row; each C-tile row lives in one 16-lane half -> shfl_xor 1,2,4,8 */
#pragma unroll 1
    for (int mt = 0; mt < 4; ++mt) {
#pragma unroll
      for (int g = 0; g < 8; ++g) {
        int i   = mt * 16 + g + mh;
        int icl = i < NTOK ? i : (NTOK - 1);
        int yi  = icl / 7, xi = icl - yi * 7;
        float vals[4];
        float mx = -1e30f;
#pragma unroll
        for (int nt = 0; nt < 4; ++nt) {
          int j = nt * 16 + l15;
          float sv;
          if (i >= NTOK) sv = 0.0f;            /* pad rows: uniform, harmless */
          else if (j >= NTOK) sv = -1e30f;     /* mask pad cols */
          else {
            int yj  = j / 7, xj = j - yj * 7;
            int rel = (yi - yj + 6) * 13 + (xi - xj + 6);
            sv = sacc[mt][nt][g] + smem[BIAS_OFF + rel * HEADS + wv];
          }
          vals[nt] = sv;
          mx = fmaxf(mx, sv);
        }
#pragma unroll
        for (int d = 1; d < 16; d <<= 1) mx = fmaxf(mx, __shfl_xor(mx, d, 32));
        float sum = 0.f;
#pragma unroll
        for (int nt = 0; nt < 4; ++nt) { vals[nt] = __expf(vals[nt] - mx); sum += vals[nt]; }
#pragma unroll
        for (int d = 1; d < 16; d <<= 1) sum += __shfl_xor(sum, d, 32);
        float inv = 1.0f / sum;
#pragma unroll
        for (int nt = 0; nt < 4; ++nt) sacc[mt][nt][g] = vals[nt] * inv;
      }
    }
  }
  __syncthreads();  /* all waves done reading q/k -> tok+q region reusable */

  /* ---- Stage 3: stage P, O = P @ V, write attn-out (64x256) to LDS ---- */
  {
    float* stg = smem + wv * STG_SZ;  /* per-wave slice of dead tok+q region */
#pragma unroll
    for (int mt = 0; mt < 4; ++mt)
#pragma unroll
      for (int nt = 0; nt < 4; ++nt)
#pragma unroll
        for (int g = 0; g < 8; ++g) {
          int m = mt * 16 + g + mh;
          stg[m * STG_S + nt * 16 + l15] = sacc[mt][nt][g];
        }

    const float* vh = smem + V_OFF + wv * (NP * QKV_S);
    v8f oacc[4][2];
#pragma unroll
    for (int mt = 0; mt < 4; ++mt)
#pragma unroll
      for (int nt = 0; nt < 2; ++nt) oacc[mt][nt] = vzero;

#pragma unroll 1
    for (int kk = 0; kk < NP; kk += 4) {
      v2f a[4], b[2];
#pragma unroll
      for (int mt = 0; mt < 4; ++mt) a[mt] = fragA(stg, STG_S, mt * 16, kk, l15, khh);
#pragma unroll
      for (int nt = 0; nt < 2; ++nt) b[nt] = fragB_KN(vh, QKV_S, kk, nt * 16, l15, khh);
#pragma unroll
      for (int mt = 0; mt < 4; ++mt)
#pragma unroll
        for (int nt = 0; nt < 2; ++nt)
          WMMA_F32(oacc[mt][nt], a[mt], b[nt]);
    }

    float* ao = smem + AO_OFF;
#pragma unroll
    for (int mt = 0; mt < 4; ++mt)
#pragma unroll
      for (int nt = 0; nt < 2; ++nt)
#pragma unroll
        for (int g = 0; g < 8; ++g) {
          int m  = mt * 16 + g + mh;
          int cc = wv * HD + nt * 16 + l15;
          ao[m * AO_S + cc] = oacc[mt][nt][g];
        }
  }
  __syncthreads();

  /* ---- Stage 4: final = attn_out(64x256) @ Wout(256x256); wave owns 32 cols ---- */
  {
    const float* ao = smem + AO_OFF;
    const int nb = wv * HD;
    v8f facc[4][2];
#pragma unroll
    for (int mt = 0; mt < 4; ++mt)
#pragma unroll
      for (int nt = 0; nt < 2; ++nt) facc[mt][nt] = vzero;

#pragma unroll 1
    for (int kk = 0; kk < DIM; kk += 4) {
      v2f a[4], b[2];
#pragma unroll
      for (int mt = 0; mt < 4; ++mt) a[mt] = fragA(ao, AO_S, mt * 16, kk, l15, khh);
#pragma unroll
      for (int nt = 0; nt < 2; ++nt) b[nt] = fragB_KN(Wout, DIM, kk, nb + nt * 16, l15, khh);
#pragma unroll
      for (int mt = 0; mt < 4; ++mt)
#pragma unroll
        for (int nt = 0; nt < 2; ++nt)
          WMMA_F32(facc[mt][nt], a[mt], b[nt]);
    }

    float* op = out + (size_t)win * WINE;
#pragma unroll
    for (int mt = 0; mt < 4; ++mt)
#pragma unroll
      for (int nt = 0; nt < 2; ++nt)
#pragma unroll
        for (int g = 0; g < 8; ++g) {
          int m = mt * 16 + g + mh;
          if (m < NTOK)
            op[m * DIM + nb + nt * 16 + l15] = facc[mt][nt][g];
        }
  }
}

extern "C" void kernel_launch(void* const* d_in, const int* in_sizes, int n_in,
                              void* d_out, int out_size, void* d_ws, size_t ws_size,
                              hipStream_t stream) {
  const float* x    = (const float*)d_in[0];
  const float* Wqkv = (const float*)d_in[1];
  const float* Wout = (const float*)d_in[2];
  const float* bt   = (const float*)d_in[3];
  float* out        = (float*)d_out;

  const int nwin = in_sizes[0] / WINE;           /* 2048 windows */
  const size_t lds_bytes = (size_t)LDS_FLOATS * sizeof(float);  /* 273952 B */
  (void)hipFuncSetAttribute((const void*)win_attn_fused,
                            hipFuncAttributeMaxDynamicSharedMemorySize,
                            (int)lds_bytes);

  hipLaunchKernelGGL(win_attn_fused, dim3(nwin), dim3(256), lds_bytes, stream,
                     x, Wqkv, Wout, bt, out);
}